// QLayer_65481071409305
// MI455X (gfx1250) — compile-verified
//
#include <hip/hip_runtime.h>

// MI455X / gfx1250 quantum-layer kernel.
// State of 8 batch elements (16 real columns: re/im) lives entirely in LDS
// (~298 KB of the WGP's 320 KB). RY layers -> 16x16 kron matrices applied with
// chained V_WMMA_F32_16X16X4_F32. CNOT chains -> row-permutation table.

typedef __attribute__((ext_vector_type(2))) float v2f;
typedef __attribute__((ext_vector_type(8))) float v8f;

#define NQ       12
#define NSTATE   4096
#define DEPTH    4
#define BPW      8      // batch elements per workgroup
#define NCOL     16     // columns = BPW * {re, im}
#define NTHREADS 256    // 8 waves of 32

// Composite CNOT-chain permutation: value at dest k comes from src sigma(k).
// chain order q=0..10 applied to state => pull back sigma_10 first.
__device__ __forceinline__ unsigned sigma_chain(unsigned k) {
#pragma unroll
  for (int q = 10; q >= 0; --q) {
    unsigned c = 11u - (unsigned)q;   // control = qubit q  -> bit 11-q
    unsigned t = 10u - (unsigned)q;   // target  = qubit q+1-> bit 10-q
    k ^= ((k >> c) & 1u) << t;
  }
  return k;
}

// Logical state row for group G (compile-time), tile t (0..255), index i.
// Group G owns k-bits [11-4G .. 8-4G]; i = (k >> (8-4G)) & 15.
template <int G>
__device__ __forceinline__ unsigned row_of(unsigned t, unsigned i) {
  if constexpr (G == 0) return (i << 8) | t;
  else if constexpr (G == 1) return ((t & 0xF0u) << 4) | (i << 4) | (t & 0x0Fu);
  else return (t << 4) | i;
}

// Apply one 16x16 group matrix (Mg, row-major in LDS) to all 256 tiles of the
// state, through the logical->physical row map Mcur. Straight-line, uniform
// control flow: EXEC is all-ones around every WMMA.
template <int G>
__device__ __forceinline__ void apply_group(const float* __restrict__ Mg,
                                            float* __restrict__ S,
                                            const unsigned short* __restrict__ Mcur,
                                            int wave, int hf, int colc) {
  // A operand (16x4 chunks of Mg): M = colc, K = v + 2*hf within chunk j
  v2f a0, a1, a2, a3;
  a0.x = Mg[colc * 16 +  0 + 2 * hf];  a0.y = Mg[colc * 16 +  1 + 2 * hf];
  a1.x = Mg[colc * 16 +  4 + 2 * hf];  a1.y = Mg[colc * 16 +  5 + 2 * hf];
  a2.x = Mg[colc * 16 +  8 + 2 * hf];  a2.y = Mg[colc * 16 +  9 + 2 * hf];
  a3.x = Mg[colc * 16 + 12 + 2 * hf];  a3.y = Mg[colc * 16 + 13 + 2 * hf];

  for (int t = wave; t < 256; t += 8) {
#define PHYS(i) ((int)Mcur[row_of<G>((unsigned)t, (unsigned)(i))])
    // B operand (4x16 chunks of state tile): N = colc, K = v + 2*hf
    v2f b0, b1, b2, b3;
    b0.x = S[PHYS( 0 + 2 * hf) * NCOL + colc];
    b0.y = S[PHYS( 1 + 2 * hf) * NCOL + colc];
    b1.x = S[PHYS( 4 + 2 * hf) * NCOL + colc];
    b1.y = S[PHYS( 5 + 2 * hf) * NCOL + colc];
    b2.x = S[PHYS( 8 + 2 * hf) * NCOL + colc];
    b2.y = S[PHYS( 9 + 2 * hf) * NCOL + colc];
    b3.x = S[PHYS(12 + 2 * hf) * NCOL + colc];
    b3.y = S[PHYS(13 + 2 * hf) * NCOL + colc];

    v8f acc = {0.f, 0.f, 0.f, 0.f, 0.f, 0.f, 0.f, 0.f};
    acc = __builtin_amdgcn_wmma_f32_16x16x4_f32(false, a0, false, b0, (short)0, acc, false, false);
    acc = __builtin_amdgcn_wmma_f32_16x16x4_f32(false, a1, false, b1, (short)0, acc, false, false);
    acc = __builtin_amdgcn_wmma_f32_16x16x4_f32(false, a2, false, b2, (short)0, acc, false, false);
    acc = __builtin_amdgcn_wmma_f32_16x16x4_f32(false, a3, false, b3, (short)0, acc, false, false);

    // D layout: M = v + 8*hf, N = colc. Tiles are disjoint -> in-place OK.
#pragma unroll
    for (int v = 0; v < 8; ++v)
      S[PHYS(v + 8 * hf) * NCOL + colc] = acc[v];
#undef PHYS
  }
}

__global__ __launch_bounds__(NTHREADS, 1)
void qlayer_kernel(const float* __restrict__ x,
                   const float* __restrict__ rys,
                   float* __restrict__ out) {
  __shared__ float          S[NSTATE * NCOL];        // 256 KB state (rows x cols)
  __shared__ unsigned short Mmap[2][NSTATE];         // 16 KB logical->physical row maps
  __shared__ float          mats[DEPTH * 3 * 256];   // 12 KB kron gate matrices
  __shared__ float          red[NCOL * NQ * 16];     // 12 KB reduction scratch
  __shared__ float          pz[NCOL * NQ];           // per-column z sums
  __shared__ float          csC[BPW * NQ], csS[BPW * NQ];   // RX half-angle trig
  __shared__ float          ryC[DEPTH * NQ], ryS[DEPTH * NQ];

  const int tid  = threadIdx.x;
  const int wave = tid >> 5;
  const int lane = tid & 31;
  const int hf   = lane >> 4;    // which half of the wave (row-dim split)
  const int colc = lane & 15;    // column (N) / A-matrix row (M)

  // ---- phase 0: trig tables + identity row map -----------------------------
  if (tid < BPW * NQ) {
    int b = tid / NQ, q = tid % NQ;
    float th = 0.5f * x[(blockIdx.x * BPW + b) * NQ + q];
    csC[tid] = cosf(th);
    csS[tid] = sinf(th);
  }
  if (tid < DEPTH * NQ) {
    float th = 0.5f * rys[tid];         // rys is (DEPTH, NQ) row-major
    ryC[tid] = cosf(th);
    ryS[tid] = sinf(th);
  }
  for (int k = tid; k < NSTATE; k += NTHREADS)
    Mmap[0][k] = (unsigned short)k;
  __syncthreads();

  // ---- phase 1: build 16x16 kron RY matrices (per depth, per qubit group) --
  for (int e = tid; e < DEPTH * 3 * 256; e += NTHREADS) {
    int mid = e >> 8;                   // d*3 + g
    int d = mid / 3, g = mid % 3;
    int i = (e >> 4) & 15, j = e & 15;
    float prod = 1.0f;
#pragma unroll
    for (int t = 0; t < 4; ++t) {
      int qi = (i >> (3 - t)) & 1;
      int qj = (j >> (3 - t)) & 1;
      int qq = d * NQ + (g * 4 + t);
      float c = ryC[qq], s = ryS[qq];
      // RY = [[c, -s], [s, c]]
      float f = qi ? (qj ? c : s) : (qj ? -s : c);
      prod *= f;
    }
    mats[e] = prod;
  }

  // ---- phase 1b: analytic RX product state ---------------------------------
  // amp(k) = prod_q (bit ? sin : cos) * (-i)^popcount(k)
  for (int k = tid; k < NSTATE; k += NTHREADS) {
    int p = __popc(k) & 3;
#pragma unroll
    for (int b = 0; b < BPW; ++b) {
      float r = 1.0f;
#pragma unroll
      for (int q = 0; q < NQ; ++q)
        r *= ((k >> (11 - q)) & 1) ? csS[b * NQ + q] : csC[b * NQ + q];
      float re = (p == 0) ? r : ((p == 2) ? -r : 0.0f);
      float im = (p == 1) ? -r : ((p == 3) ? r : 0.0f);
      S[k * NCOL + 2 * b]     = re;
      S[k * NCOL + 2 * b + 1] = im;
    }
  }
  __syncthreads();

  // ---- phase 2: depth loop: 3 grouped-RY GEMMs + CNOT map compose ----------
  int cur = 0;
#pragma unroll
  for (int d = 0; d < DEPTH; ++d) {
    const unsigned short* Mcur = Mmap[cur];
    apply_group<0>(&mats[(d * 3 + 0) * 256], S, Mcur, wave, hf, colc);
    __syncthreads();
    apply_group<1>(&mats[(d * 3 + 1) * 256], S, Mcur, wave, hf, colc);
    __syncthreads();
    apply_group<2>(&mats[(d * 3 + 2) * 256], S, Mcur, wave, hf, colc);
    __syncthreads();
    // Compose CNOT chain: M_{d+1}[k] = M_d[sigma(k)]  (ping-pong buffers)
    for (int k = tid; k < NSTATE; k += NTHREADS)
      Mmap[1 - cur][k] = Mmap[cur][sigma_chain((unsigned)k)];
    cur ^= 1;
    __syncthreads();
  }
  // cur == 0 here; Mmap[0] = sigma^DEPTH (final logical->physical map)

  // ---- phase 3: z_q = sum_k sign_q(k) * (re^2 + im^2), deterministic -------
  {
    int c = tid & 15;        // column
    int chunk = tid >> 4;    // 16 chunks of 256 logical rows
    float acc[NQ];
#pragma unroll
    for (int q = 0; q < NQ; ++q) acc[q] = 0.0f;
    for (int k = chunk * 256; k < chunk * 256 + 256; ++k) {
      float v = S[(int)Mmap[cur][k] * NCOL + c];
      float v2 = v * v;
#pragma unroll
      for (int q = 0; q < NQ; ++q)
        acc[q] += ((k >> (11 - q)) & 1) ? -v2 : v2;
    }
#pragma unroll
    for (int q = 0; q < NQ; ++q)
      red[(c * NQ + q) * 16 + chunk] = acc[q];
  }
  __syncthreads();
  if (tid < NCOL * NQ) {
    float s = 0.0f;
#pragma unroll
    for (int ch = 0; ch < 16; ++ch) s += red[tid * 16 + ch];
    pz[tid] = s;
  }
  __syncthreads();
  if (tid < BPW * NQ) {
    int b = tid / NQ, q = tid % NQ;
    // combine re column (2b) and im column (2b+1)
    out[(blockIdx.x * BPW + b) * NQ + q] =
        pz[(2 * b) * NQ + q] + pz[(2 * b + 1) * NQ + q];
  }
}

extern "C" void kernel_launch(void* const* d_in, const int* in_sizes, int n_in,
                              void* d_out, int out_size, void* d_ws, size_t ws_size,
                              hipStream_t stream) {
  (void)n_in; (void)out_size; (void)d_ws; (void)ws_size;
  const float* x   = (const float*)d_in[0];   // (B, 12)
  const float* rys = (const float*)d_in[1];   // (4, 12)
  // d_in[2] (cnot_params) is unused by the reference computation.
  float* out = (float*)d_out;                 // (B, 12)
  int B   = in_sizes[0] / NQ;                 // 512
  int nwg = B / BPW;                          // 64 workgroups
  qlayer_kernel<<<nwg, NTHREADS, 0, stream>>>(x, rys, out);
}